// GCNSyntheticPerturb_synthetic_78219944394950
// MI455X (gfx1250) — compile-verified
//
#include <hip/hip_runtime.h>
#include <hip/hip_bf16.h>
#include <math.h>

// ---------------------------------------------------------------------------
// GCNSyntheticPerturb forward, fp32, CDNA5 (gfx1250).
//  - A_sig = sigmoid(P_tril_symm) * sub_adj materialized once (64MB, L2-resident)
//    with column sums fused.
//  - Symmetric normalization (A_sig + I) * dinv_i * dinv_j fused into the WMMA
//    GEMM's A-tile staging.
//  - fp32 WMMA GEMM (v_wmma_f32_16x16x4_f32): LDS double-buffered (global
//    loads of tile k+1 overlap WMMA on tile k, one barrier/iter), vectorized
//    LDS staging (ds_store_b128 for A, interleaved b64 pairs for B), K-paired
//    B layout so every fragment is a single ds_load_b64. K-split + atomic
//    accumulation spreads the big 4096x4096 spmm across WGPs.
// ---------------------------------------------------------------------------

#define NV    4096      // nodes
#define NF    512       // features
#define NH    60        // hidden
#define NHP   64        // hidden padded
#define NC    4         // classes

typedef __attribute__((ext_vector_type(2))) float v2f;
typedef __attribute__((ext_vector_type(8))) float v8f;

// ---------------- adjacency pipeline ----------------

__global__ void k_init_d(float* __restrict__ d, int n) {
  int j = blockIdx.x * 256 + threadIdx.x;
  if (j < n) d[j] = 1.0f;                           // self-loop contribution
}

// Fused: A_sig[i,j] = sigmoid(P_hat_symm[i,j]) * sub_adj[i,j], plus column
// partial sums (one atomicAdd per column per 128-row strip).
__global__ void __launch_bounds__(256) k_build_colsum(
    const float* __restrict__ P, const float* __restrict__ adj,
    float* __restrict__ Asig, float* __restrict__ d, int n)
{
  const int j  = blockIdx.x * 256 + threadIdx.x;
  const int i0 = blockIdx.y * 128;
  float s = 0.0f;
  for (int i = i0; i < i0 + 128; ++i) {
    int hi = i > j ? i : j;
    int lo = i > j ? j : i;
    float p   = P[(size_t)hi * (hi + 1) / 2 + lo];  // tril scatter + symmetrize
    float sig = 1.0f / (1.0f + expf(-p));
    float a   = sig * adj[(size_t)i * n + j];
    Asig[(size_t)i * n + j] = a;
    s += a;
  }
  atomicAdd(&d[j], s);
}

__global__ void k_dinv(const float* __restrict__ d, float* __restrict__ dinv, int n) {
  int j = blockIdx.x * 256 + threadIdx.x;
  if (j >= n) return;
  float v = d[j];
  dinv[j] = (v > 0.0f) ? 1.0f / sqrtf(v) : 0.0f;
}

// zero-pad a rows x cols weight into prows x pcols
__global__ void k_pad(const float* __restrict__ W, float* __restrict__ Wp,
                      int rows, int cols, int prows, int pcols)
{
  int idx = blockIdx.x * 256 + threadIdx.x;
  if (idx >= prows * pcols) return;
  int r = idx / pcols, c = idx - r * pcols;
  Wp[idx] = (r < rows && c < cols) ? W[r * cols + c] : 0.0f;
}

__global__ void k_zero(float* __restrict__ p, int n) {
  int i = blockIdx.x * 256 + threadIdx.x;
  if (i < n) p[i] = 0.0f;
}

// bias + optional relu epilogue for the K-split (atomic) GEMM path
__global__ void k_bias_act(float* __restrict__ C, const float* __restrict__ bias,
                           int bias_n, int do_relu, int n)
{
  int i = blockIdx.x * 256 + threadIdx.x;
  if (i >= n) return;
  int col = i & (NHP - 1);
  float v = C[i] + (col < bias_n ? bias[col] : 0.0f);
  if (do_relu) v = fmaxf(v, 0.0f);
  C[i] = v;
}

// ---------------- fp32 WMMA GEMM: C[M x 64] (+)= A[M x K] @ B[K x 64] -------
// BM=128 rows/block, BN=64 (full width), BK=32. 256 threads = 8 waves; wave w
// owns rows [w*16, w*16+16) and 4 accumulators. If dscale != nullptr, A is the
// raw A_sig and we form (A + I) * dscale_i * dscale_j during staging.
// gridDim.y > 1 => K-split partial blocks atomically accumulate into C
// (caller zeroes C and applies bias/relu in a separate pass).

#define GBM 128
#define GBK 32
#define GAS 36       // A LDS stride: 16B-aligned rows + conflict-free b64 reads
#define GBS 160      // B LDS pair-row stride; 160 % 64 == 32 => halves disjoint

__global__ void __launch_bounds__(256) k_gemm_wmma(
    const float* __restrict__ A, int lda, int K,
    const float* __restrict__ B,                // K x 64 row-major
    float* __restrict__ C,                      // M x 64 row-major
    const float* __restrict__ bias, int bias_n, int do_relu,
    const float* __restrict__ dscale)
{
  __shared__ float As[2][GBM * GAS];            // 2 x 18432 B
  __shared__ float Bs[2][(GBK / 2) * GBS];      // 2 x 10240 B, K-paired layout

  const int tid  = threadIdx.x;
  const int lane = tid & 31;
  const int wave = tid >> 5;        // 0..7
  const int half = lane >> 4;       // 0: lanes 0-15, 1: lanes 16-31
  const int l16  = lane & 15;
  const int rowBlock = blockIdx.x * GBM;

  const int kchunk = K / gridDim.y;
  const int kbeg   = blockIdx.y * kchunk;
  const int kend   = kbeg + kchunk;

  v8f acc[4] = {};                  // 4 tiles of 16x16 fp32

  const int a_c4 = (tid & 7) * 4;   // A-tile: 8 threads per 32-wide row
  const int a_r  = tid >> 3;        // 0..31, +32 per pass (4 passes = 128 rows)
  const int b_q  = tid >> 4;        // 0..15: K row-pair owned by this thread
  const int b_c4 = (tid & 15) * 4;  // 16 threads cover a 64-wide row pair

  float4 ra[4];                     // in-flight A tile (global -> regs)
  float4 rb[2];                     // in-flight B row pair

  auto load_regs = [&](int k0) {
#pragma unroll
    for (int p = 0; p < 4; ++p) {
      int r  = a_r + p * 32;
      int gr = rowBlock + r;
      const float* src = &A[(size_t)gr * lda + k0 + a_c4];
      float4 v = *(const float4*)src;
      if (dscale) {                 // fused (A + I) * dinv_i * dinv_j
        float  di = dscale[gr];
        float4 dj = *(const float4*)&dscale[k0 + a_c4];
        int    gc = k0 + a_c4;
        v.x = (v.x + ((gc + 0 == gr) ? 1.0f : 0.0f)) * di * dj.x;
        v.y = (v.y + ((gc + 1 == gr) ? 1.0f : 0.0f)) * di * dj.y;
        v.z = (v.z + ((gc + 2 == gr) ? 1.0f : 0.0f)) * di * dj.z;
        v.w = (v.w + ((gc + 3 == gr) ? 1.0f : 0.0f)) * di * dj.w;
      }
      ra[p] = v;
      if (k0 + GBK < kend) __builtin_prefetch(src + GBK, 0, 3);  // global_prefetch_b8
    }
    rb[0] = *(const float4*)&B[(size_t)(k0 + 2 * b_q + 0) * NHP + b_c4];
    rb[1] = *(const float4*)&B[(size_t)(k0 + 2 * b_q + 1) * NHP + b_c4];
  };

  auto store_lds = [&](int buf) {
#pragma unroll
    for (int p = 0; p < 4; ++p)     // ds_store_b128, 16B-aligned (GAS % 4 == 0)
      *(float4*)&As[buf][(a_r + p * 32) * GAS + a_c4] = ra[p];
    const float* s0 = (const float*)&rb[0];
    const float* s1 = (const float*)&rb[1];
#pragma unroll
    for (int m = 0; m < 4; ++m) {   // interleave K-pair: ds_store_b64
      v2f bp;
      bp.x = s0[m];
      bp.y = s1[m];
      *(v2f*)&Bs[buf][b_q * GBS + (b_c4 + m) * 2] = bp;
    }
  };

  // prologue: stage first tile
  load_regs(kbeg);
  store_lds(0);
  __syncthreads();

  int buf = 0;
  for (int k0 = kbeg; k0 < kend; k0 += GBK) {
    const bool has_next = (k0 + GBK) < kend;
    if (has_next) load_regs(k0 + GBK);        // global loads overlap WMMA below

#pragma unroll
    for (int kk = 0; kk < GBK; kk += 4) {
      // A fragment (16x4): lanes 0-15 hold K=kk,kk+1 ; lanes 16-31 K=kk+2,kk+3
      const v2f a = *(const v2f*)&As[buf][(wave * 16 + l16) * GAS + kk + half * 2];
      const float* brow = &Bs[buf][((kk >> 1) + half) * GBS];
#pragma unroll
      for (int t = 0; t < 4; ++t) {
        const v2f b = *(const v2f*)&brow[(t * 16 + l16) * 2];  // single b64
        acc[t] = __builtin_amdgcn_wmma_f32_16x16x4_f32(
            false, a, false, b, (short)0, acc[t], false, false);
      }
    }

    if (has_next) store_lds(buf ^ 1);         // fill other buffer
    __syncthreads();                          // one barrier per K-tile
    buf ^= 1;
  }

  // ---- epilogue: C/D layout — VGPR r: lanes 0-15 -> base+r, 16-31 -> base+8+r
  const int row0 = rowBlock + wave * 16 + half * 8;
  if (gridDim.y == 1) {
#pragma unroll
    for (int t = 0; t < 4; ++t) {
      const int col  = t * 16 + l16;
      const float bv = (bias && col < bias_n) ? bias[col] : 0.0f;
#pragma unroll
      for (int r = 0; r < 8; ++r) {
        float v = acc[t][r] + bv;
        if (do_relu) v = fmaxf(v, 0.0f);
        C[(size_t)(row0 + r) * NHP + col] = v;
      }
    }
  } else {
#pragma unroll
    for (int t = 0; t < 4; ++t) {
      const int col = t * 16 + l16;
#pragma unroll
      for (int r = 0; r < 8; ++r)
        atomicAdd(&C[(size_t)(row0 + r) * NHP + col], acc[t][r]);
    }
  }
}

// ---------------- head: logits = H3[:, :60] @ lin_w + lin_b; log_softmax ----
__global__ void __launch_bounds__(256) k_head(
    const float* __restrict__ H3, const float* __restrict__ lw,
    const float* __restrict__ lb, float* __restrict__ out, int n)
{
  int r = blockIdx.x * 256 + threadIdx.x;
  if (r >= n) return;
  float acc[NC] = {lb[0], lb[1], lb[2], lb[3]};
  const float* h = &H3[(size_t)r * NHP];
  for (int c = 0; c < NH; ++c) {
    float hv = h[c];
#pragma unroll
    for (int k = 0; k < NC; ++k) acc[k] += hv * lw[c * NC + k];
  }
  float m = fmaxf(fmaxf(acc[0], acc[1]), fmaxf(acc[2], acc[3]));
  float s = 0.0f;
#pragma unroll
  for (int k = 0; k < NC; ++k) s += expf(acc[k] - m);
  float lse = m + logf(s);
#pragma unroll
  for (int k = 0; k < NC; ++k) out[(size_t)r * NC + k] = acc[k] - lse;
}

// ---------------------------------------------------------------------------

extern "C" void kernel_launch(void* const* d_in, const int* in_sizes, int n_in,
                              void* d_out, int out_size, void* d_ws, size_t ws_size,
                              hipStream_t stream) {
  const float* x    = (const float*)d_in[0];    // 4096 x 512
  const float* adj  = (const float*)d_in[1];    // 4096 x 4096
  const float* Pvec = (const float*)d_in[2];    // N*(N+1)/2
  const float* W1   = (const float*)d_in[3];    // 512 x 60
  const float* b1   = (const float*)d_in[4];
  const float* W2   = (const float*)d_in[5];    // 60 x 60
  const float* b2   = (const float*)d_in[6];
  const float* W3   = (const float*)d_in[7];    // 60 x 60
  const float* b3   = (const float*)d_in[8];
  const float* lw   = (const float*)d_in[9];    // 60 x 4
  const float* lb   = (const float*)d_in[10];
  float* out = (float*)d_out;

  // workspace carve-up (floats)
  float* ws   = (float*)d_ws;
  float* Asig = ws;                               size_t off = (size_t)NV * NV;
  float* dsum = ws + off;                         off += NV;
  float* dinv = ws + off;                         off += NV;
  float* F0   = ws + off;                         off += (size_t)NV * NHP;
  float* F1   = ws + off;                         off += (size_t)NV * NHP;
  float* W1p  = ws + off;                         off += (size_t)NF * NHP;
  float* W2p  = ws + off;                         off += (size_t)NHP * NHP;
  float* W3p  = ws + off;                         off += (size_t)NHP * NHP;

  const int FELEMS = NV * NHP;                    // 262144
  const int FBLK   = FELEMS / 256;

  // 0) pad weights to WMMA-tile-aligned shapes
  k_pad<<<(NF * NHP) / 256, 256, 0, stream>>>(W1, W1p, NF, NH, NF, NHP);
  k_pad<<<(NHP * NHP) / 256, 256, 0, stream>>>(W2, W2p, NH, NH, NHP, NHP);
  k_pad<<<(NHP * NHP) / 256, 256, 0, stream>>>(W3, W3p, NH, NH, NHP, NHP);

  // 1) adjacency: fused build + column sums, then d^-1/2
  k_init_d<<<NV / 256, 256, 0, stream>>>(dsum, NV);
  k_build_colsum<<<dim3(NV / 256, NV / 128), 256, 0, stream>>>(Pvec, adj, Asig, dsum, NV);
  k_dinv<<<NV / 256, 256, 0, stream>>>(dsum, dinv, NV);

  const dim3 gDense(NV / GBM, 1);                 // feature GEMMs: direct store
  const dim3 gSpmm (NV / GBM, 8);                 // spmm: K-split x8, atomic acc

  // 2) layer 1
  k_gemm_wmma<<<gDense, 256, 0, stream>>>(x, NF, NF, W1p, F0, nullptr, 0, 0, nullptr);
  k_zero<<<FBLK, 256, 0, stream>>>(F1, FELEMS);
  k_gemm_wmma<<<gSpmm, 256, 0, stream>>>(Asig, NV, NV, F0, F1, nullptr, 0, 0, dinv);
  k_bias_act<<<FBLK, 256, 0, stream>>>(F1, b1, NH, 1, FELEMS);
  // 3) layer 2
  k_gemm_wmma<<<gDense, 256, 0, stream>>>(F1, NHP, NHP, W2p, F0, nullptr, 0, 0, nullptr);
  k_zero<<<FBLK, 256, 0, stream>>>(F1, FELEMS);
  k_gemm_wmma<<<gSpmm, 256, 0, stream>>>(Asig, NV, NV, F0, F1, nullptr, 0, 0, dinv);
  k_bias_act<<<FBLK, 256, 0, stream>>>(F1, b2, NH, 1, FELEMS);
  // 4) layer 3 (no relu)
  k_gemm_wmma<<<gDense, 256, 0, stream>>>(F1, NHP, NHP, W3p, F0, nullptr, 0, 0, nullptr);
  k_zero<<<FBLK, 256, 0, stream>>>(F1, FELEMS);
  k_gemm_wmma<<<gSpmm, 256, 0, stream>>>(Asig, NV, NV, F0, F1, nullptr, 0, 0, dinv);
  k_bias_act<<<FBLK, 256, 0, stream>>>(F1, b3, NH, 0, FELEMS);

  // 5) head + log_softmax
  k_head<<<NV / 256, 256, 0, stream>>>(F1, lw, lb, out, NV);
}